// Af_14027363188859
// MI455X (gfx1250) — compile-verified
//
#include <hip/hip_runtime.h>

// Bilinear affine warp (grid_sample-style), b=16 f=16 H=304 W=608, fp32.
// Memory-bound: ~380 MB min HBM traffic -> ~16us @ 23.3 TB/s. No matrix ops
// (6 FMA/pixel affine is fused inline; WMMA would add traffic, not remove it).
// CDNA5 paths used: NT temporal-hint stores (protect 192MB L2 holding the
// gather footprint), clustered 16-deep global_load gathers per thread,
// scalar-cache loads for the per-image affine.

typedef float v4f __attribute__((ext_vector_type(4)));

#define W_DIM 608
#define H_DIM 304
#define HW_DIM (W_DIM * H_DIM)      // 184832
#define N_IMG  256                  // b*f
#define QUADS  (HW_DIM / 4)         // 46208 (W divisible by 4 -> quads row-aligned)

__global__ __launch_bounds__(256) void
warp_bilinear_kernel(const float* __restrict__ stim,
                     const float* __restrict__ eye,
                     float* __restrict__ out)
{
    const int img = blockIdx.y;                       // (b,f) pair
    const int q   = blockIdx.x * blockDim.x + threadIdx.x;
    if (q >= QUADS) return;

    // Per-image affine: uniform across block -> scalar loads via K$.
    const float* __restrict__ E = eye + img * 6;
    const float a00 = E[0], a01 = E[1], a02 = E[2];
    const float a10 = E[3], a11 = E[4], a12 = E[5];

    const int   p0   = q * 4;
    const int   row  = p0 / W_DIM;                    // const-divide -> mul/shift
    const int   col0 = p0 - row * W_DIM;

    // jnp.linspace(-1,1,N): step = 2/(N-1)
    const float yg = fmaf((float)row, 2.0f / (H_DIM - 1), -1.0f);
    // Fold the yg and constant terms once per quad.
    const float bx = fmaf(a01, yg, a02);
    const float by = fmaf(a11, yg, a12);

    const float* __restrict__ im = stim + (size_t)img * HW_DIM;

    float res[4];
#pragma unroll
    for (int k = 0; k < 4; ++k) {
        const float xg = fmaf((float)(col0 + k), 2.0f / (W_DIM - 1), -1.0f);
        const float xn = fmaf(a00, xg, bx);           // affine x (normalized)
        const float yn = fmaf(a10, xg, by);           // affine y (normalized)
        const float x  = (xn + 1.0f) * (0.5f * (float)W_DIM);
        const float y  = (yn + 1.0f) * (0.5f * (float)H_DIM);

        int x0 = (int)floorf(x);
        int y0 = (int)floorf(y);
        int x1 = x0 + 1;
        int y1 = y0 + 1;
        // clamp (reference clamps indices, then derives weights from clamped)
        x0 = x0 < 0 ? 0 : (x0 > W_DIM - 1 ? W_DIM - 1 : x0);
        x1 = x1 < 0 ? 0 : (x1 > W_DIM - 1 ? W_DIM - 1 : x1);
        y0 = y0 < 0 ? 0 : (y0 > H_DIM - 1 ? H_DIM - 1 : y0);
        y1 = y1 < 0 ? 0 : (y1 > H_DIM - 1 ? H_DIM - 1 : y1);

        // 4 independent gathers; unrolled x4 -> 16 outstanding loads/thread.
        const float Ia = im[y0 * W_DIM + x0];
        const float Ib = im[y1 * W_DIM + x0];
        const float Ic = im[y0 * W_DIM + x1];
        const float Id = im[y1 * W_DIM + x1];

        const float x0f = (float)x0, x1f = (float)x1;
        const float y0f = (float)y0, y1f = (float)y1;
        const float wa = (x1f - x) * (y1f - y);
        const float wb = (x1f - x) * (y - y0f);
        const float wc = (x - x0f) * (y1f - y);
        const float wd = (x - x0f) * (y - y0f);

        res[k] = fmaf(wa, Ia, fmaf(wb, Ib, fmaf(wc, Ic, wd * Id)));
    }

    // Write-once stream: non-temporal 128-bit store, keep L2 for the gathers.
    v4f v = { res[0], res[1], res[2], res[3] };
    __builtin_nontemporal_store(v, (v4f*)(out + (size_t)img * HW_DIM + p0));
}

extern "C" void kernel_launch(void* const* d_in, const int* in_sizes, int n_in,
                              void* d_out, int out_size, void* d_ws, size_t ws_size,
                              hipStream_t stream)
{
    const float* stim = (const float*)d_in[0];   // (16,16,304,608) f32
    const float* eye  = (const float*)d_in[1];   // (16,16,6)       f32
    float*       out  = (float*)d_out;           // (16,16,304,608) f32

    dim3 block(256);                              // 8 wave32s
    dim3 grid((QUADS + 255) / 256, N_IMG);        // (181, 256)
    warp_bilinear_kernel<<<grid, block, 0, stream>>>(stim, eye, out);
}